// GraphConvolution_sparse_996432412814
// MI455X (gfx1250) — compile-verified
//
#include <hip/hip_runtime.h>

#define F 128  // F_IN == F_OUT == 128

typedef __attribute__((ext_vector_type(2))) float v2f;
typedef __attribute__((ext_vector_type(4))) float v4f;
typedef __attribute__((ext_vector_type(8))) float v8f;

// ---------------------------------------------------------------------------
// Kernel 1: support = X @ W  using V_WMMA_F32_16X16X4_F32 (full fp32 WMMA).
// grid.x = ceil(N/16) row tiles. Block = 256 threads = 8 waves.
// Wave w computes the 16x16 output tile at (row0, 16*w). K loop = 128/4 = 32.
//
// VGPR layouts (ISA 7.12.2, wave32):
//   A 16x4 f32 : lane = (half<<4)|m ; a[j] holds A[m][2*half + j], j=0,1
//   B 4x16 f32 : b[j] holds B[2*half + j][m]           (mirrors A)
//   C/D 16x16  : c[j] holds C[j + 8*half][m], j=0..7
// ---------------------------------------------------------------------------
__global__ __launch_bounds__(256) void gemm_wmma_f32(
    const float* __restrict__ X, const float* __restrict__ W,
    float* __restrict__ S, int N)
{
  const int wave = threadIdx.x >> 5;
  const int lane = threadIdx.x & 31;
  const int half = lane >> 4;   // 0: lanes 0-15, 1: lanes 16-31
  const int m    = lane & 15;
  const int row0 = blockIdx.x * 16;
  const int col0 = wave * 16;

  // Clamp A-row reads for a ragged last tile; stores are guarded below.
  int arow = row0 + m;
  if (arow >= N) arow = N - 1;

  const float* __restrict__ xrow = X + (size_t)arow * F;  // A: one row per lane
  const float* __restrict__ wcol = W + col0 + m;          // B: one col per lane

  v8f c = {};
#pragma unroll
  for (int k = 0; k < F; k += 4) {
    v2f a;
    a.x = xrow[k + 2 * half + 0];
    a.y = xrow[k + 2 * half + 1];
    v2f b;
    b.x = wcol[(size_t)(k + 2 * half + 0) * F];
    b.y = wcol[(size_t)(k + 2 * half + 1) * F];
    // (neg_a, A, neg_b, B, c_mod, C, reuse_a, reuse_b)
    c = __builtin_amdgcn_wmma_f32_16x16x4_f32(false, a, false, b,
                                              (short)0, c, false, false);
  }

  float* __restrict__ srow = S + (size_t)row0 * F + col0;
#pragma unroll
  for (int j = 0; j < 8; ++j) {
    const int r = j + 8 * half;
    if (row0 + r < N) srow[(size_t)r * F + m] = c[j];
  }
}

// ---------------------------------------------------------------------------
// Kernel 2: out[n][f] = bias[f]  (vectorized b128 broadcast)
// total4 = N*F/4 ; bias4 has 32 entries (128 floats).
// ---------------------------------------------------------------------------
__global__ __launch_bounds__(256) void init_bias(
    v4f* __restrict__ out4, const v4f* __restrict__ bias4, int total4)
{
  const int i = blockIdx.x * 256 + threadIdx.x;
  if (i < total4) out4[i] = bias4[i & 31];
}

// ---------------------------------------------------------------------------
// Kernel 3: out[u] += support[v] per edge. One wave per edge:
// lane l loads support[v][4l..4l+3] as b128 (512B/row per wave, L2-resident)
// and issues 4 native global_atomic_add_f32 into out[u].
// ---------------------------------------------------------------------------
__global__ __launch_bounds__(256) void scatter_add(
    const int* __restrict__ eidx, const float* __restrict__ S,
    float* __restrict__ out, int E)
{
  const int wave = threadIdx.x >> 5;
  const int lane = threadIdx.x & 31;
  const int e = blockIdx.x * 8 + wave;
  if (e >= E) return;

  const int u = eidx[e];       // destination (segment id)
  const int v = eidx[E + e];   // source node

  const v4f val = ((const v4f*)(S + (size_t)v * F))[lane];
  float* __restrict__ dst = out + (size_t)u * F + lane * 4;

  __hip_atomic_fetch_add(dst + 0, val.x, __ATOMIC_RELAXED, __HIP_MEMORY_SCOPE_AGENT);
  __hip_atomic_fetch_add(dst + 1, val.y, __ATOMIC_RELAXED, __HIP_MEMORY_SCOPE_AGENT);
  __hip_atomic_fetch_add(dst + 2, val.z, __ATOMIC_RELAXED, __HIP_MEMORY_SCOPE_AGENT);
  __hip_atomic_fetch_add(dst + 3, val.w, __ATOMIC_RELAXED, __HIP_MEMORY_SCOPE_AGENT);
}

// ---------------------------------------------------------------------------
extern "C" void kernel_launch(void* const* d_in, const int* in_sizes, int n_in,
                              void* d_out, int out_size, void* d_ws, size_t ws_size,
                              hipStream_t stream)
{
  const float* X    = (const float*)d_in[0];  // [N, 128]
  const int*   eidx = (const int*)d_in[1];    // [2, E] (u row, then v row)
  const float* W    = (const float*)d_in[2];  // [128, 128]
  const float* bias = (const float*)d_in[3];  // [128]
  float* out = (float*)d_out;                 // [N, 128]
  float* S   = (float*)d_ws;                  // support scratch: N*128 f32

  const int N = in_sizes[0] / F;   // 50000
  const int E = in_sizes[1] / 2;   // 640000

  // 1) support = X @ W   (WMMA f32, memory-bound)
  gemm_wmma_f32<<<(N + 15) / 16, 256, 0, stream>>>(X, W, S, N);

  // 2) out = bias broadcast
  const int total4 = (N * F) / 4;
  init_bias<<<(total4 + 255) / 256, 256, 0, stream>>>(
      (v4f*)out, (const v4f*)bias, total4);

  // 3) out[u] += support[v]   (L2-resident atomics)
  scatter_add<<<(E + 7) / 8, 256, 0, stream>>>(eidx, S, out, E);
}